// LGCNEncoder_39797166964864
// MI455X (gfx1250) — compile-verified
//
#include <hip/hip_runtime.h>

#define N_USERS 100000
#define N_ITEMS 50000
#define NN      150000          // total nodes
#define E       128             // embedding size
#define NNZ_E   4800000
#define BATCH   4096
#define NLAYERS 3

typedef __attribute__((ext_vector_type(2)))  float  v2f;
typedef __attribute__((ext_vector_type(8)))  float  v8f;

// ---------------------------------------------------------------------------
// Zero the pos/dis accumulators (2 * NN * E floats), float4-wide.
// ---------------------------------------------------------------------------
__global__ void zero_f4(float4* __restrict__ p, int n4) {
  int i = blockIdx.x * blockDim.x + threadIdx.x;
  if (i < n4) p[i] = make_float4(0.f, 0.f, 0.f, 0.f);
}

// ---------------------------------------------------------------------------
// Edge-parallel SpMM scatter: one wave32 per edge, 4 floats per lane.
//   y[row[e], :] += val[e] * x[col[e], :]
// x/y (76.8MB each) are kept L2-resident; edge lists are streamed with
// non-temporal loads so they don't evict the gather/scatter working set.
// Atomics resolve in L2 (f32 hardware atomics via unsafeAtomicAdd).
// ---------------------------------------------------------------------------
__global__ void spmm_scatter(const int*   __restrict__ row,
                             const int*   __restrict__ col,
                             const float* __restrict__ val,
                             const float* __restrict__ x,
                             float*       __restrict__ y) {
  int e = (blockIdx.x * blockDim.x + threadIdx.x) >> 5;
  if (e >= NNZ_E) return;
  int lane = threadIdx.x & 31;
  int   r = __builtin_nontemporal_load(row + e);   // TH=NT: stream edge lists
  int   c = __builtin_nontemporal_load(col + e);
  float v = __builtin_nontemporal_load(val + e);
  const float4* xr = (const float4*)(x + (size_t)c * E);
  float4 xv = xr[lane];
  float* yr = y + (size_t)r * E + lane * 4;
  unsafeAtomicAdd(yr + 0, v * xv.x);
  unsafeAtomicAdd(yr + 1, v * xv.y);
  unsafeAtomicAdd(yr + 2, v * xv.z);
  unsafeAtomicAdd(yr + 3, v * xv.w);
}

// ---------------------------------------------------------------------------
// ego[N,128] = concat(pos, dis)[N,256] @ Wk[256,128] + bk[128]
//
// One 256-thread block (8 waves) per 16-row strip:
//   - stage the 16x256 fp32 X-tile into LDS once (coalesced b128 loads),
//     row-padded to 260 floats so the 16-lane row-strided A reads are
//     bank-conflict-free (260 mod 64 = 4 -> distinct banks per lane)
//   - each wave computes one 16x16 tile with V_WMMA_F32_16X16X4_F32
//     (exact fp32 reference precision), A from LDS, B from L2-resident W
// ---------------------------------------------------------------------------
#define XPAD 260

__global__ void gemm_wmma_bias(const float* __restrict__ pos,
                               const float* __restrict__ dis,
                               const float* __restrict__ Wk,   // [256,128]
                               const float* __restrict__ bk,   // [128]
                               float*       __restrict__ out)  // [NN,128]
{
  __shared__ float Xtile[16][XPAD];

  int tid   = threadIdx.x;          // 0..255
  int lane  = tid & 31;
  int wv    = tid >> 5;             // 0..7
  int mBase = blockIdx.x * 16;
  int half  = lane >> 4;            // 0 or 1
  int l15   = lane & 15;
  int nCol  = wv * 16 + l15;        // 0..127

  // ---- cooperative load of X-tile: 16 rows x 256 cols = 1024 float4 ----
  #pragma unroll
  for (int p = 0; p < 4; ++p) {
    int li  = p * 256 + tid;        // 0..1023
    int r   = li >> 6;              // 0..15 (64 float4 per row)
    int c4  = li & 63;
    int c   = c4 * 4;               // 0..252
    const float* src = (c < 128)
        ? (pos + (size_t)(mBase + r) * E + c)
        : (dis + (size_t)(mBase + r) * E + (c - 128));
    float4 vv = *(const float4*)src;
    *(float4*)&Xtile[r][c] = vv;
  }
  __syncthreads();

  // ---- WMMA K-loop: A 16x4 from LDS, B 4x16 from global (L2-resident) ----
  v8f acc = {};
  int kh = half * 2;                // K offset pair for this lane half
  #pragma unroll 4
  for (int kb = 0; kb < 256; kb += 4) {
    v2f a = *(const v2f*)&Xtile[l15][kb + kh];     // ds_load_b64, no conflicts
    v2f bm;
    bm.x = Wk[(size_t)(kb + kh) * E + nCol];
    bm.y = Wk[(size_t)(kb + kh + 1) * E + nCol];
    acc = __builtin_amdgcn_wmma_f32_16x16x4_f32(
        /*neg_a=*/false, a, /*neg_b=*/false, bm,
        /*c_mod=*/(short)0, acc, /*reuse_a=*/false, /*reuse_b=*/false);
  }

  // D layout: acc[r] -> (M = mBase + half*8 + r, N = nCol); fuse bias.
  float bv = bk[nCol];
  int mOut = mBase + half * 8;
  #pragma unroll
  for (int r = 0; r < 8; ++r)
    out[(size_t)(mOut + r) * E + nCol] = acc[r] + bv;
}

// ---------------------------------------------------------------------------
// out[0:4096]      = ego[users[i]]
// out[4096:8192]   = ego[N_USERS + items[i]]
// ---------------------------------------------------------------------------
__global__ void gather_out(const float* __restrict__ ego,
                           const int*   __restrict__ users,
                           const int*   __restrict__ items,
                           float*       __restrict__ out) {
  int i = blockIdx.x;   // 0..8191
  int j = threadIdx.x;  // 0..127
  size_t src;
  if (i < BATCH) src = (size_t)users[i] * E + j;
  else           src = (size_t)(N_USERS + items[i - BATCH]) * E + j;
  out[(size_t)i * E + j] = ego[src];
}

// ---------------------------------------------------------------------------
extern "C" void kernel_launch(void* const* d_in, const int* in_sizes, int n_in,
                              void* d_out, int out_size, void* d_ws, size_t ws_size,
                              hipStream_t stream) {
  const float* user_emb = (const float*)d_in[0];
  const float* item_emb = (const float*)d_in[1];
  const float* adj_val  = (const float*)d_in[2];
  const float* hp_val   = (const float*)d_in[3];
  const float* W        = (const float*)d_in[4];   // [3,256,128]
  const float* bvec     = (const float*)d_in[5];   // [3,128]
  const int*   adj_row  = (const int*)d_in[6];
  const int*   adj_col  = (const int*)d_in[7];
  const int*   hp_row   = (const int*)d_in[8];
  const int*   hp_col   = (const int*)d_in[9];
  const int*   users    = (const int*)d_in[10];
  const int*   items    = (const int*)d_in[11];
  float* out = (float*)d_out;

  const size_t NE = (size_t)NN * E;                // 19,200,000 floats
  float* pos = (float*)d_ws;
  float* dis = pos + NE;
  float* ego = dis + NE;

  // ego = concat(user_emb, item_emb)   (capturable d2d copies)
  hipMemcpyAsync(ego, user_emb, (size_t)N_USERS * E * sizeof(float),
                 hipMemcpyDeviceToDevice, stream);
  hipMemcpyAsync(ego + (size_t)N_USERS * E, item_emb,
                 (size_t)N_ITEMS * E * sizeof(float),
                 hipMemcpyDeviceToDevice, stream);

  const int zeroN4   = (int)(2 * NE / 4);                        // pos+dis as float4
  const int zeroGrid = (zeroN4 + 255) / 256;
  const int spmmGrid = (NNZ_E + 7) / 8;                          // 8 edges/block (256 thr)
  const int gemmGrid = NN / 16;                                  // 9375 strips

  for (int k = 0; k < NLAYERS; ++k) {
    zero_f4<<<zeroGrid, 256, 0, stream>>>((float4*)pos, zeroN4);
    spmm_scatter<<<spmmGrid, 256, 0, stream>>>(adj_row, adj_col, adj_val, ego, pos);
    spmm_scatter<<<spmmGrid, 256, 0, stream>>>(hp_row, hp_col, hp_val, ego, dis);
    gemm_wmma_bias<<<gemmGrid, 256, 0, stream>>>(
        pos, dis, W + (size_t)k * 256 * E, bvec + (size_t)k * E, ego);
  }

  gather_out<<<2 * BATCH, E, 0, stream>>>(ego, users, items, out);
}